// PDFAHyberbolicParameterization_16982300688422
// MI455X (gfx1250) — compile-verified
//
#include <hip/hip_runtime.h>
#include <math.h>

// ---------------------------------------------------------------------------
// PDFA hyperbolic parameterization — MI455X (gfx1250) implementation.
//
// Math: out[b] = logsumexp_n( log( (init @ Prod_l M_{b,l})[n] + 1e-20 ) + accL[n] )
//   where M_{b,l}[p,n] = sum_c Tn[p,c,n] * exp(logx[b,l,c]),
//         Tn = hyperbolic_softmax(invh_T, axis=n), init = hyperbolic_softmax(invh_init),
//         accL = log(hsig(invh_acc)).
//
// Strategy:
//   prep_kernel    : normalize T / init / acc into workspace (tiny).
//   pdfa_chunk     : one wave32 per (batch, chunk). Keeps the running 16x16
//                    product P in WMMA C/D layout and iterates the chunk in
//                    REVERSE order:  P <- M_l x P, implemented as four
//                    V_WMMA_F32_16X16X4_F32 (A = column-slices of M_l,
//                    B = row-slices of P).  M_l is generated directly in
//                    A-operand layout from a per-lane register cache of T
//                    (64 floats/lane); P's row-slices are rebuilt from the
//                    D layout with 8 ds_bpermute + 8 cndmask per step.
//   pdfa_finish    : per batch, fold row-vector init through the chunk
//                    products, then logsumexp.  Negligible cost.
// ---------------------------------------------------------------------------

constexpr int S = 16;   // states
constexpr int C = 8;    // characters
constexpr int B = 64;   // batch
constexpr int L = 4096; // sequence length

typedef float v2f __attribute__((ext_vector_type(2)));
typedef float v8f __attribute__((ext_vector_type(8)));

// workspace layout (in floats)
constexpr int WS_T     = 0;      // S*C*S = 2048 : normalized transition probs
constexpr int WS_INIT  = 2048;   // 16          : initial state probs
constexpr int WS_ACC   = 2064;   // 16          : accepting-state log probs
constexpr int WS_CHUNK = 2080;   // B*nchunk*256: per-chunk 16x16 products

__device__ __forceinline__ float hsig(float x) {
  return x / (1.0f + fabsf(x)) * 0.5f + 0.5f;
}

__device__ __forceinline__ float bperm(int idx_bytes, float v) {
  return __int_as_float(__builtin_amdgcn_ds_bpermute(idx_bytes, __float_as_int(v)));
}

// ---------------------------------------------------------------------------
// Prep: 1 block x 128 threads; thread t handles transition row (p,c) = t.
// ---------------------------------------------------------------------------
__global__ void __launch_bounds__(128)
pdfa_prep_kernel(const float* __restrict__ invh_init,
                 const float* __restrict__ invh_T,
                 const float* __restrict__ invh_acc,
                 float* __restrict__ ws) {
  const int t = threadIdx.x;  // 0..127 == p*C + c
  float vals[S];
  float sum = 0.0f;
#pragma unroll
  for (int n = 0; n < S; ++n) {
    float h = hsig(invh_T[t * S + n]);
    vals[n] = h;
    sum += h;
  }
  const float inv = 1.0f / sum;
#pragma unroll
  for (int n = 0; n < S; ++n) ws[WS_T + t * S + n] = vals[n] * inv;

  if (t < S) {
    float s2 = 0.0f;
#pragma unroll
    for (int p = 0; p < S; ++p) s2 += hsig(invh_init[p]);
    ws[WS_INIT + t] = hsig(invh_init[t]) / s2;
    ws[WS_ACC + t]  = logf(hsig(invh_acc[t]));
  }
}

// ---------------------------------------------------------------------------
// Phase 1: one wave32 per (batch, chunk).  Chunk product of CHUNKLEN matrices.
//
// Layouts (wave32, per CDNA5 ISA 7.12.2), lane = m + 16*hi (m = lane&15):
//   A 16x4 f32  : vgpr v holds A[m, v + 2*hi]                (rows across lanes)
//   B 4x16 f32  : vgpr v holds B[v + 2*hi, m]   (inferred by analogy w/ 8-bit B)
//   C/D 16x16   : vgpr r holds D[r + 8*hi, m]
// ---------------------------------------------------------------------------
template <int CHUNKLEN>
__global__ void __launch_bounds__(32)
pdfa_chunk_kernel(const float* __restrict__ logx,
                  const float* __restrict__ ws,
                  int nchunk) {
  __shared__ float xs[CHUNKLEN * C];

  const int lane = threadIdx.x;
  const int b    = blockIdx.x / nchunk;
  const int k    = blockIdx.x % nchunk;
  const int m    = lane & 15;
  const int hi   = lane >> 4;

  // Stage exp(logx) for this chunk into LDS (contiguous CHUNKLEN*C floats).
  const float* src = logx + ((size_t)b * L + (size_t)k * CHUNKLEN) * C;
  for (int i = lane; i < CHUNKLEN * C; i += 32) xs[i] = __expf(src[i]);

  // Per-lane T cache: Treg[2*j+v][c] = Tn[m][c][4*j + v + 2*hi]
  // (exactly the entries this lane contributes to A-operand vgpr v of slice j)
  float Treg[8][8];
  const float* T = ws + WS_T;
#pragma unroll
  for (int j = 0; j < 4; ++j)
#pragma unroll
    for (int v = 0; v < 2; ++v) {
      const int n = 4 * j + v + 2 * hi;
#pragma unroll
      for (int c = 0; c < 8; ++c) Treg[2 * j + v][c] = T[(m * C + c) * S + n];
    }

  __syncthreads();

  // Running product P = identity, in C/D layout: acc[r] = P[r + 8*hi][m].
  v8f acc;
#pragma unroll
  for (int r = 0; r < 8; ++r) acc[r] = ((r + 8 * hi) == m) ? 1.0f : 0.0f;

  const int idxLo = m * 4;         // ds_bpermute byte index: read lane m
  const int idxHi = (16 + m) * 4;  // read lane 16+m

  for (int step = CHUNKLEN - 1; step >= 0; --step) {
    // x[c] = exp(logx[b, l, c]) — broadcast across the wave from LDS.
    float x[8];
#pragma unroll
    for (int c = 0; c < 8; ++c) x[c] = xs[step * C + c];

    // Build A = M_l in A layout:  a[2j+v] = sum_c Treg[2j+v][c] * x[c]
    float a[8];
#pragma unroll
    for (int q = 0; q < 8; ++q) {
      float s = Treg[q][0] * x[0];
#pragma unroll
      for (int c = 1; c < 8; ++c) s = fmaf(Treg[q][c], x[c], s);
      a[q] = s;
    }

    // Build B slices of P from D layout.  bb[2j+v] must hold P[4j+v+2hi][m].
    // Row r' of P lives in acc[r'&7], lane-half (r'>>3).
    float bb[8];
    {
      float s;
      // j=0: rows 0(lo),2(hi) | 1(lo),3(hi)   — sources in lo lane half
      s = bperm(idxLo, acc[2]); bb[0] = hi ? s : acc[0];
      s = bperm(idxLo, acc[3]); bb[1] = hi ? s : acc[1];
      // j=1: rows 4(lo),6(hi) | 5(lo),7(hi)
      s = bperm(idxLo, acc[6]); bb[2] = hi ? s : acc[4];
      s = bperm(idxLo, acc[7]); bb[3] = hi ? s : acc[5];
      // j=2: rows 8(lo),10(hi) | 9(lo),11(hi) — sources in hi lane half
      s = bperm(idxHi, acc[0]); bb[4] = hi ? acc[2] : s;
      s = bperm(idxHi, acc[1]); bb[5] = hi ? acc[3] : s;
      // j=3: rows 12(lo),14(hi) | 13(lo),15(hi)
      s = bperm(idxHi, acc[4]); bb[6] = hi ? acc[6] : s;
      s = bperm(idxHi, acc[5]); bb[7] = hi ? acc[7] : s;
    }

    // P_new = M_l x P = sum_j A_j x B_j  via four V_WMMA_F32_16X16X4_F32.
    v8f d;
#pragma unroll
    for (int r = 0; r < 8; ++r) d[r] = 0.0f;
#pragma unroll
    for (int j = 0; j < 4; ++j) {
      v2f Av = {a[2 * j], a[2 * j + 1]};
      v2f Bv = {bb[2 * j], bb[2 * j + 1]};
      d = __builtin_amdgcn_wmma_f32_16x16x4_f32(
          /*neg_a=*/false, Av, /*neg_b=*/false, Bv,
          /*c_mod=*/(short)0, d, /*reuse_a=*/false, /*reuse_b=*/false);
    }
    acc = d;
  }

  // Store chunk product row-major: out[row][col], row = r + 8*hi, col = m.
  float* dst = const_cast<float*>(ws) + WS_CHUNK + ((size_t)b * nchunk + k) * 256;
#pragma unroll
  for (int r = 0; r < 8; ++r) dst[(r + 8 * hi) * 16 + m] = acc[r];
}

// ---------------------------------------------------------------------------
// Phase 2: one wave per batch.  v <- init; v <- v @ C_k; then logsumexp.
// Lane holds component n = lane & 15 (hi half mirrors lo half harmlessly).
// ---------------------------------------------------------------------------
__global__ void __launch_bounds__(32)
pdfa_finish_kernel(const float* __restrict__ ws, float* __restrict__ out,
                   int nchunk) {
  const int b = blockIdx.x;
  const int lane = threadIdx.x;
  const int n = lane & 15;

  float v = ws[WS_INIT + n];
  const float* chunks = ws + WS_CHUNK + (size_t)b * nchunk * 256;
  for (int k = 0; k < nchunk; ++k) {
    const float* M = chunks + k * 256;
    float nv = 0.0f;
#pragma unroll
    for (int p = 0; p < 16; ++p) {
      float vp = __shfl(v, p, 32);         // broadcast v[p] from lane p
      nv = fmaf(vp, M[p * 16 + n], nv);
    }
    v = nv;
  }

  // logsumexp over n of log(v + 1e-20) + accL[n]
  float s = logf(v + 1e-20f) + ws[WS_ACC + n];
  float mx = s;
#pragma unroll
  for (int off = 8; off >= 1; off >>= 1) mx = fmaxf(mx, __shfl_xor(mx, off, 16));
  float ex = __expf(s - mx);
#pragma unroll
  for (int off = 8; off >= 1; off >>= 1) ex += __shfl_xor(ex, off, 16);
  if (lane == 0) out[b] = mx + logf(ex);
}

// ---------------------------------------------------------------------------
extern "C" void kernel_launch(void* const* d_in, const int* in_sizes, int n_in,
                              void* d_out, int out_size, void* d_ws,
                              size_t ws_size, hipStream_t stream) {
  const float* logx      = (const float*)d_in[0];  // (B, L, C)
  const float* invh_init = (const float*)d_in[1];  // (S,)
  const float* invh_T    = (const float*)d_in[2];  // (S, C, S)
  const float* invh_acc  = (const float*)d_in[3];  // (S,)
  float* ws  = (float*)d_ws;
  float* out = (float*)d_out;                      // (1, B) -> B floats

  auto need_bytes = [](int nchunk) -> size_t {
    return (size_t)(WS_CHUNK + B * nchunk * 256) * sizeof(float);
  };
  int nchunk;
  if      (ws_size >= need_bytes(64)) nchunk = 64;  // 4096 waves, 64-step chains
  else if (ws_size >= need_bytes(16)) nchunk = 16;  // smaller scratch fallback
  else                                nchunk = 4;   // minimal scratch fallback

  pdfa_prep_kernel<<<1, 128, 0, stream>>>(invh_init, invh_T, invh_acc, ws);

  dim3 grid(B * nchunk);
  switch (nchunk) {
    case 64: pdfa_chunk_kernel<L / 64><<<grid, 32, 0, stream>>>(logx, ws, 64); break;
    case 16: pdfa_chunk_kernel<L / 16><<<grid, 32, 0, stream>>>(logx, ws, 16); break;
    default: pdfa_chunk_kernel<L / 4 ><<<grid, 32, 0, stream>>>(logx, ws, 4);  break;
  }

  pdfa_finish_kernel<<<B, 32, 0, stream>>>(ws, out, nchunk);
}